// AerialPatchSampler_32057635898124
// MI455X (gfx1250) — compile-verified
//
#include <hip/hip_runtime.h>
#include <cstdint>
#include <cstddef>

// AerialPatchSampler for MI455X (gfx1250).
// Per (pose, 32-channel half) block: DMA the rotated patch's bounding box
// into LDS with the CDNA5 Tensor Data Mover (one tensor_load_to_lds per
// 4-channel tile, TENSORcnt-tracked, double-buffered so DMA overlaps the
// bilinear compute), then do all 4 taps as LDS reads. Output is streamed
// with non-temporal stores to keep L2 (192 MB) for the feature planes.
// Memory-bound op: ~512 MB HBM round trip -> ~22 us floor at 23.3 TB/s.

#define C_TOT   64
#define HA_     512
#define WA_     512
#define HB_     32
#define WB_     32
#define CT      4                    // channels per LDS tile
#define CHALF   32                   // channels per block
#define NTILES  (CHALF / CT)         // 8
#define MAXROWS 46                   // max bbox rows: 31*sqrt(2)+2 < 46
#define MAXW    52                   // max bbox width in dwords (13 x 4-dword chunks)
#define BUF_DW  (CT * MAXROWS * MAXW)        // 9568 dwords per buffer
#define SMEM_DW (2 * BUF_DW + 64)            // double buffer + clamped-tap pad

#if defined(__HIP_DEVICE_COMPILE__) && __has_builtin(__builtin_amdgcn_tensor_load_to_lds)
#define USE_TDM 1
#else
#define USE_TDM 0
#endif

typedef unsigned int v4u __attribute__((ext_vector_type(4)));
typedef int          v8i __attribute__((ext_vector_type(8)));
typedef int          v4i __attribute__((ext_vector_type(4)));

__global__ __launch_bounds__(256)
void AerialPatchSampler_gfx1250_kernel(const float* __restrict__ aer,
                                       const float* __restrict__ pose,
                                       float* __restrict__ out)
{
    __shared__ __align__(16) float smem[SMEM_DW];

    const int tid   = threadIdx.x;
    const int bn    = blockIdx.x;            // b*256 + n
    const int b     = bn >> 8;
    const int chOff = blockIdx.y * CHALF;    // 0 or 32

    // Pose (uniform per block; FEATURE_SCALE==1 => ix = gu, iy = gv exactly).
    const float u = pose[bn * 3 + 0];
    const float v = pose[bn * 3 + 1];
    const float r = pose[bn * 3 + 2];
    // r in [-pi, pi]: hardware sin/cos (v_sin_f32/v_cos_f32), no slow path.
    const float cth =  __cosf(r);            // cos(-r)
    const float sth = -__sinf(r);            // sin(-r)

    // Bounding box of ix = u + cth*u0 - sth*v0, iy = v + sth*u0 + cth*v0
    // over u0 in [0,31], v0 in [-16,15] (linear => extremes at corners).
    const float a31c = 31.0f * cth, a31s = 31.0f * sth;
    const float ixmin = u + fminf(0.0f, a31c) + fminf( 16.0f * sth, -15.0f * sth);
    const float ixmax = u + fmaxf(0.0f, a31c) + fmaxf( 16.0f * sth, -15.0f * sth);
    const float iymin = v + fminf(0.0f, a31s) + fminf(-16.0f * cth,  15.0f * cth);
    const float iymax = v + fmaxf(0.0f, a31s) + fmaxf(-16.0f * cth,  15.0f * cth);

    const int x_lo = (int)floorf(ixmin);
    const int x_hi = (int)floorf(ixmax) + 1;       // +1 for the x1 bilinear corner
    const int y_lo = (int)floorf(iymin);
    const int y_hi = (int)floorf(iymax) + 1;

    // Clamp to image; 4-dword-aligned x origin so DMA rows (512-dword pitch)
    // never run past the buffer end.
    const int x0a = (min(max(x_lo, 0), WA_ - 4)) & ~3;
    const int x1c =  min(max(x_hi, 0), WA_ - 1);
    const int y0c =  min(max(y_lo, 0), HA_ - 1);
    const int y1c =  min(max(y_hi, 0), HA_ - 1);
    int w4 = (x1c >= x0a) ? (((x1c - x0a) >> 2) + 1) : 1;
    int Hl = (y1c >= y0c) ? ((y1c - y0c) + 1) : 1;
    w4 = min(w4, MAXW / 4);
    Hl = min(Hl, MAXROWS);
    const int W = w4 << 2;           // LDS row pitch (dwords) == loaded width
    const int S = Hl * W;            // LDS dwords per channel

    // Per-pixel geometry (channel-invariant): 4 pixels per thread.
    float fx_[4], fy_[4];
    int   off_[4];
    bool  v00_[4], v01_[4], v10_[4], v11_[4];
#pragma unroll
    for (int q = 0; q < 4; ++q) {
        const int   p  = tid + (q << 8);
        const int   i  = p >> 5, j = p & 31;
        const float u0 = (float)(HB_ - 1 - i);
        const float v0 = (float)(j - (WB_ >> 1));
        const float ix = fmaf(cth, u0, fmaf(-sth, v0, u));
        const float iy = fmaf(sth, u0, fmaf( cth, v0, v));
        // |gx|<1 && |gy|<1  <=>  -0.5 < ix,iy < 511.5
        const bool pv = (ix > -0.5f) && (ix < 511.5f) && (iy > -0.5f) && (iy < 511.5f);
        const float xf = floorf(ix), yf = floorf(iy);
        const int   x0 = (int)xf,    y0 = (int)yf;
        fx_[q] = ix - xf;
        fy_[q] = iy - yf;
        const bool vx0 = pv && (x0 >= 0)     && (x0 < WA_);
        const bool vx1 = pv && (x0 + 1 >= 0) && (x0 + 1 < WA_);
        const bool vy0 = pv && (y0 >= 0)     && (y0 < HA_);
        const bool vy1 = pv && (y0 + 1 >= 0) && (y0 + 1 < HA_);
        v00_[q] = vx0 && vy0;  v01_[q] = vx1 && vy0;
        v10_[q] = vx0 && vy1;  v11_[q] = vx1 && vy1;
        // Clamped LDS coords: never distort a *valid* corner (valid => in box),
        // invalid taps stay inside the padded allocation and are masked to 0.
        const int lx = min(max(x0 - x0a, 0), W - 1);
        const int ly = min(max(y0 - y0c, 0), Hl - 1);
        off_[q] = ly * W + lx;
    }

    const size_t plane = (size_t)HA_ * WA_;                 // 262144
    const float* tile0 = aer + (size_t)b * C_TOT * plane
                             + (size_t)chOff * plane
                             + (size_t)y0c * WA_ + x0a;
    float* outp = out + ((size_t)bn * C_TOT + chOff) * (size_t)(HB_ * WB_);

#if USE_TDM
    // Tensor DMA descriptor, invariant parts (D# groups 1/2/3).
    v8i g1 = {0, 0, 0, 0, 0, 0, 0, 0};
    g1[0] = (2 << 16);                                        // data_size = 4 bytes
    g1[1] = (int)(((unsigned)WA_ & 0xFFFFu) << 16);           // tensor_dim0 lo16
    g1[2] = (int)((((unsigned)WA_ >> 16) & 0xFFFFu)
                  | (((unsigned)HA_ & 0xFFFFu) << 16));       // dim0 hi | dim1 lo
    g1[3] = (int)((((unsigned)HA_ >> 16) & 0xFFFFu)
                  | (((unsigned)W & 0xFFFFu) << 16));         // dim1 hi | tile_dim0
    g1[4] = (int)(((unsigned)Hl & 0xFFFFu)
                  | (((unsigned)CT & 0xFFFFu) << 16));        // tile_dim1 | tile_dim2
    g1[5] = (int)WA_;                                         // tensor_dim0_stride lo32
    g1[6] = (int)(((unsigned)(plane & 0xFFFFu)) << 16);       // stride0 hi16 | stride1 lo16
    g1[7] = (int)((unsigned)(plane >> 16));                   // tensor_dim1_stride [47:16]
    v4i g2 = {0, 0, 0, 0};
    g2[0] = C_TOT;                                            // tensor_dim2 (z OOB bound)
    v4i g3 = {0, 0, 0, 0};
    const unsigned lds_base = (unsigned)(uintptr_t)(&smem[0]); // low 32 = LDS byte addr

    // Issue one 3D-tile DMA (x=W dwords, y=Hl rows, z=CT channels) for tile t
    // into ping-pong buffer `sel`.
    auto tdm_issue = [&](int t, int sel) {
        const uint64_t ga = (uint64_t)(uintptr_t)(tile0 + (size_t)(t * CT) * plane);
        v4u g0;
        g0[0] = 1u;                                               // count=1 (valid)
        g0[1] = lds_base + (unsigned)(sel * (BUF_DW * 4));        // LDS byte address
        g0[2] = (unsigned)(ga & 0xFFFFFFFFu);                     // global_addr [31:0]
        g0[3] = (unsigned)((ga >> 32) & 0x1FFFFFFu) | (2u << 30); // addr[56:32] | type=2
#if defined(__clang_major__) && __clang_major__ >= 23
        v8i g5 = {0, 0, 0, 0, 0, 0, 0, 0};
        __builtin_amdgcn_tensor_load_to_lds(g0, g1, g2, g3, g5, 0);
#else
        __builtin_amdgcn_tensor_load_to_lds(g0, g1, g2, g3, 0);
#endif
    };

    if (tid < 32) {                  // wave 0 drives the DMA pipeline
        tdm_issue(0, 0);
    }
#endif

    for (int t = 0; t < NTILES; ++t) {
#if USE_TDM
        if (tid < 32) {
            if (t + 1 < NTILES) {
                // Prefetch next tile into the other buffer (safe: the
                // end-of-iteration barrier of t-1 guarantees no wave still
                // reads it), then wait for tile t only (tensor ops complete
                // in order within a wave).
                tdm_issue(t + 1, (t + 1) & 1);
                __builtin_amdgcn_s_wait_tensorcnt(1);
            } else {
                __builtin_amdgcn_s_wait_tensorcnt(0);
            }
        }
        __syncthreads();             // tile t now visible to all waves
        const float* buf = smem + (t & 1) * BUF_DW;
#else
        __syncthreads();
        float* bufw = smem + (t & 1) * BUF_DW;
        for (int f = tid; f < CT * S; f += 256) {
            const int cc = f / S;
            const int rr = f - cc * S;
            const int y  = rr / W;
            const int x  = rr - y * W;
            bufw[f] = tile0[(size_t)(t * CT + cc) * plane + (size_t)y * WA_ + x];
        }
        __syncthreads();
        const float* buf = bufw;
#endif

        // Bilinear taps from LDS; coalesced 32-lane non-temporal stores
        // (write-once output must not evict feature planes from L2).
#pragma unroll
        for (int cc = 0; cc < CT; ++cc) {
            const float* sm = buf + cc * S;
            float* o = outp + (size_t)(t * CT + cc) * (HB_ * WB_);
#pragma unroll
            for (int q = 0; q < 4; ++q) {
                const int   off = off_[q];
                const float v00 = v00_[q] ? sm[off]         : 0.0f;
                const float v01 = v01_[q] ? sm[off + 1]     : 0.0f;
                const float v10 = v10_[q] ? sm[off + W]     : 0.0f;
                const float v11 = v11_[q] ? sm[off + W + 1] : 0.0f;
                const float fx = fx_[q], fy = fy_[q];
                const float top = fmaf(fx, v01 - v00, v00);
                const float bot = fmaf(fx, v11 - v10, v10);
                __builtin_nontemporal_store(fmaf(fy, bot - top, top),
                                            o + tid + (q << 8));
            }
        }
        __syncthreads();             // all waves done with buf[t&1] before it
                                     // becomes the prefetch target at t+1
    }
}

extern "C" void kernel_launch(void* const* d_in, const int* in_sizes, int n_in,
                              void* d_out, int out_size, void* d_ws, size_t ws_size,
                              hipStream_t stream) {
    const float* aer  = (const float*)d_in[0];   // (B, 64, 512, 512) fp32
    const float* pose = (const float*)d_in[1];   // (B, 256, 3) fp32
    float* out        = (float*)d_out;           // (B, 256, 64, 32, 32) fp32

    const int BN = in_sizes[1] / 3;              // B*N = 1024
    dim3 grid((unsigned)BN, 2, 1);               // pose x channel-half
    dim3 block(256, 1, 1);
    AerialPatchSampler_gfx1250_kernel<<<grid, block, 0, stream>>>(aer, pose, out);
}